// TotalVariation_53498112639571
// MI455X (gfx1250) — compile-verified
//
#include <hip/hip_runtime.h>
#include <hip/hip_bf16.h>

// Total variation of [32,8,3,256,256] f32 tensor -> scalar.
// HBM-bandwidth-bound: ~201 MB read, roofline ~8.6us @ 23.3 TB/s on MI455X.
// Each 256-thread block owns one 256x256 plane. Each thread walks a 4-column
// strip down 64 rows carrying the current row float4 in registers, so every
// element is loaded once as b128 (plus a strip-boundary scalar that hits L0).
// Wave32 reduction via V_WMMA_F32_16X16X4_F32 ones-trick + ds_swizzle.

typedef __attribute__((ext_vector_type(2))) float v2f;
typedef __attribute__((ext_vector_type(8))) float v8f;

#define TV_EPS 1e-6f
#define PLANE_ELEMS 65536      // 256*256
#define N_TOTAL 50331648.0f    // 32*8*3*256*256

// ds_swizzle_b32 group-of-32 XOR pattern: offset = (xor<<10) | (and=0x1f)
#define SWZ_XOR_ADD(v, m)                                                   \
    __int_as_float(__builtin_amdgcn_ds_swizzle(__float_as_int(v),           \
                                               (0x1f | ((m) << 10))))

// Reduce a per-lane accumulator across the wave32 using one
// v_wmma_f32_16x16x4_f32: A = ones(16x4), B(4x16) rows 0,1 = lane accs,
// rows 2,3 = 0 => every D column n = acc[n] + acc[n+16]. VGPR0 of D holds
// the 16 column sums (mirrored in both lane halves); finish with 4 xor
// swizzle-adds. Requires EXEC all-1s at the call site.
__device__ __forceinline__ float wave32_reduce_wmma(float acc) {
    v2f a; a[0] = 1.0f; a[1] = 1.0f;   // A: 16x4 all-ones
    v2f b; b[0] = acc;  b[1] = 0.0f;   // B: rows 0/1 = lane accs, rows 2/3 = 0
    v8f c = {};
    c = __builtin_amdgcn_wmma_f32_16x16x4_f32(
            /*neg_a=*/false, a, /*neg_b=*/false, b,
            /*c_mod=*/(short)0, c, /*reuse_a=*/false, /*reuse_b=*/false);
    float v = c[0];                    // colsum[lane & 15] in every lane
    v += SWZ_XOR_ADD(v, 1);
    v += SWZ_XOR_ADD(v, 2);
    v += SWZ_XOR_ADD(v, 4);
    v += SWZ_XOR_ADD(v, 8);
    return v;                          // wave total in every lane
}

__device__ __forceinline__ void block_reduce_store(float acc, float* dst,
                                                   float scale) {
    const float wsum = wave32_reduce_wmma(acc);
    __shared__ float smem[8];
    if ((threadIdx.x & 31) == 0) smem[threadIdx.x >> 5] = wsum;
    __syncthreads();
    if (threadIdx.x == 0) {
        float s = 0.0f;
#pragma unroll
        for (int i = 0; i < 8; ++i) s += smem[i];
        dst[0] = s * scale;
    }
}

__global__ __launch_bounds__(256) void tv_plane_kernel(
        const float* __restrict__ x, float* __restrict__ partials) {
    const int plane = blockIdx.x;             // 0..767
    const int strip = threadIdx.x & 63;       // float4-wide column strip
    const int rblk  = threadIdx.x >> 6;       // 0..3, 64 rows each
    const bool has_right = (strip != 63);

    const float* p = x + (size_t)plane * PLANE_ELEMS
                       + ((size_t)rblk << 14)           // rblk*64*256
                       + ((size_t)strip << 2);

    float4 cur = *reinterpret_cast<const float4*>(p);   // row r0
    float acc = 0.0f;

    // Rows r0 .. r0+62: H-diffs of cur, W-diffs vs. next row, roll registers.
#pragma unroll 4
    for (int i = 0; i < 63; ++i) {
        const float4 dn = *reinterpret_cast<const float4*>(p + 256);
        acc += fabsf(cur.y - cur.x + TV_EPS);
        acc += fabsf(cur.z - cur.y + TV_EPS);
        acc += fabsf(cur.w - cur.z + TV_EPS);
        if (has_right) {                       // strip-boundary H-diff
            acc += fabsf(p[4] - cur.w + TV_EPS);
        }
        acc += fabsf(dn.x - cur.x + TV_EPS);
        acc += fabsf(dn.y - cur.y + TV_EPS);
        acc += fabsf(dn.z - cur.z + TV_EPS);
        acc += fabsf(dn.w - cur.w + TV_EPS);
        cur = dn;
        p += 256;
    }

    // Last row of this 64-row block (r = r0+63): H-diffs always; W-diff
    // only if a next row exists inside the plane (rblk < 3).
    acc += fabsf(cur.y - cur.x + TV_EPS);
    acc += fabsf(cur.z - cur.y + TV_EPS);
    acc += fabsf(cur.w - cur.z + TV_EPS);
    if (has_right) {
        acc += fabsf(p[4] - cur.w + TV_EPS);
    }
    if (rblk < 3) {
        const float4 dn = *reinterpret_cast<const float4*>(p + 256);
        acc += fabsf(dn.x - cur.x + TV_EPS);
        acc += fabsf(dn.y - cur.y + TV_EPS);
        acc += fabsf(dn.z - cur.z + TV_EPS);
        acc += fabsf(dn.w - cur.w + TV_EPS);
    }

    // All 256 threads re-converged here -> EXEC all-1s for WMMA.
    block_reduce_store(acc, partials + blockIdx.x, 1.0f);
}

__global__ __launch_bounds__(256) void tv_finalize_kernel(
        const float* __restrict__ partials, int n,
        float* __restrict__ out, float inv_n) {
    float acc = 0.0f;
    // n == 768 == 3 * blockDim.x: uniform loop, no divergence before WMMA.
    for (int i = threadIdx.x; i < n; i += 256) acc += partials[i];
    block_reduce_store(acc, out, inv_n);
}

extern "C" void kernel_launch(void* const* d_in, const int* in_sizes, int n_in,
                              void* d_out, int out_size, void* d_ws, size_t ws_size,
                              hipStream_t stream) {
    const float* x   = (const float*)d_in[0];
    float* out       = (float*)d_out;
    float* partials  = (float*)d_ws;                 // 768 floats of scratch

    const int planes = in_sizes[0] >> 16;            // elems / 65536 = 768

    tv_plane_kernel<<<planes, 256, 0, stream>>>(x, partials);
    tv_finalize_kernel<<<1, 256, 0, stream>>>(partials, planes, out,
                                              1.0f / N_TOTAL);
}